// RegionNet_CLIP_66855460929920
// MI455X (gfx1250) — compile-verified
//
#include <hip/hip_runtime.h>
#include <hip/hip_bf16.h>

typedef __attribute__((ext_vector_type(2))) float v2f;
typedef __attribute__((ext_vector_type(8))) float v8f;

// ---------------------------------------------------------------------------
// Stage 1: region scores -> normalized -> perturbed-argmax histogram weights
// grid = 8 (one block per b*n_key row), block = 128 threads
// ---------------------------------------------------------------------------
__global__ __launch_bounds__(128) void weights_kernel(
    const float* __restrict__ score,   // (8, 196) flat
    const float* __restrict__ noise,   // (8, 500, 121) flat
    const float* __restrict__ sigmap,  // scalar
    float* __restrict__ wout)          // (8, 121) flat
{
    __shared__ float ssc[196];
    __shared__ float sc[121];
    __shared__ float snorm[121];
    __shared__ float lohi[2];
    __shared__ int   cnt[121];

    const int bn  = blockIdx.x;
    const int tid = threadIdx.x;
    const float sigma = sigmap[0];

    for (int i = tid; i < 196; i += 128) ssc[i] = score[bn * 196 + i];
    for (int r = tid; r < 121; r += 128) cnt[r] = 0;
    __syncthreads();

    // sc[r] = mean of 4x4 window of the 14x14 score grid
    for (int r = tid; r < 121; r += 128) {
        int oy = r / 11, ox = r % 11;
        float s = 0.f;
        #pragma unroll
        for (int i = 0; i < 4; ++i)
            #pragma unroll
            for (int j = 0; j < 4; ++j)
                s += ssc[(oy + i) * 14 + (ox + j)];
        sc[r] = s * (1.0f / 16.0f);
    }
    __syncthreads();

    if (tid == 0) {
        float lo = sc[0], hi = sc[0];
        for (int r = 1; r < 121; ++r) {
            float v = sc[r];
            lo = (v < lo) ? v : lo;
            hi = (v > hi) ? v : hi;
        }
        lohi[0] = lo; lohi[1] = hi;
    }
    __syncthreads();

    const float lo = lohi[0];
    const float inv = 1.0f / (lohi[1] - lohi[0] + 1e-5f);
    for (int r = tid; r < 121; r += 128) snorm[r] = (sc[r] - lo) * inv;
    __syncthreads();

    // 500 samples: argmax over 121 of snorm[r] + noise*sigma (first-max wins)
    for (int s = tid; s < 500; s += 128) {
        const float* np = noise + (size_t)(bn * 500 + s) * 121;
        float best = -3.402823e38f;
        int bi = 0;
        #pragma unroll 1
        for (int r = 0; r < 121; ++r) {
            float v = snorm[r] + np[r] * sigma;
            if (v > best) { best = v; bi = r; }
        }
        atomicAdd(&cnt[bi], 1);
    }
    __syncthreads();

    for (int r = tid; r < 121; r += 128)
        wout[bn * 121 + r] = (float)cnt[r] * (1.0f / 500.0f);
}

// ---------------------------------------------------------------------------
// Stage 2: per (bt,c): D(16x1024) = A(16x196) x B(196x1024) via WMMA f32.
// K dimension permuted so every address is a compile-time immediate:
//   k' = kb*4 + half*2 + j, kb = 7r + s  ->  g = (2r+half, 2s+j)
//   A[q=(kh,kw)][g=(gy,gx)] = w[gy-kh, gx-kw] (0-padded 11x11), stored in LDS
//   in permuted-K order; B[g][p=(ay,ax)] = x[b,c,t, gy*32+ay, gx*32+ax]
//   D row q -> output pixel (kh*32+ay, kw*32+ax)
// grid = 16*3*8 = 384 blocks, 256 threads = 8 wave32s, one 16x16 N-tile/wave
// ---------------------------------------------------------------------------
__global__ __launch_bounds__(256) void gemm_kernel(
    const float* __restrict__ x,        // (2,3,8,448,448)
    const int*   __restrict__ group_id, // (2,8)
    const float* __restrict__ wreg,     // (8,121)
    float* __restrict__ out)            // (2,3,8,128,128)
{
    __shared__ float lA[16 * 196];      // A, row-major [m][k'] (permuted K)

    const int blk = blockIdx.x;         // 0..383
    const int bt  = blk / 24;           // 0..15
    const int rem = blk % 24;
    const int c   = rem / 8;            // 0..2
    const int ng  = rem % 8;            // n-tile group 0..7
    const int b   = bt >> 3;
    const int t   = bt & 7;
    const int tid = threadIdx.x;

    const int key = group_id[b * 8 + t];
    const float* w = wreg + (size_t)(b * 4 + key) * 121;

    // Build A in LDS, K-permuted: k' -> g = (2r+half, 2s+j)
    for (int idx = tid; idx < 16 * 196; idx += 256) {
        int q   = idx / 196;            // row m = (kh,kw)
        int kp  = idx % 196;
        int kb  = kp >> 2;
        int hlf = (kp >> 1) & 1;
        int j   = kp & 1;
        int r   = kb / 7, s = kb % 7;
        int gy  = 2 * r + hlf;
        int gx  = 2 * s + j;
        int kh  = q >> 2, kw = q & 3;
        int oy  = gy - kh, ox = gx - kw;
        float v = 0.f;
        if (oy >= 0 && oy < 11 && ox >= 0 && ox < 11) v = w[oy * 11 + ox];
        lA[idx] = v;
    }
    __syncthreads();

    const int wave = tid >> 5;          // wave32
    const int lane = tid & 31;
    const int tile = ng * 8 + wave;     // 0..63 N-tiles of 16 pixels
    const int n    = lane & 15;
    const int half = lane >> 4;         // 0/1: K sub-split per ISA layout
    const int p    = tile * 16 + n;     // pixel index 0..1023
    const int ay   = p >> 5;
    const int ax   = p & 31;
    const int m    = n;                 // A row held by this lane

    const float* xb    = x + (size_t)((b * 3 + c) * 8 + t) * (448 * 448);
    const float* baseB = xb + (half * 32 + ay) * 448 + ax;   // gy=half, gx=0, j=0
    const float* baseA = lA + m * 196 + half * 2;

    v8f acc0 = {}, acc1 = {};
    #pragma unroll
    for (int kb = 0; kb < 49; ++kb) {
        const int r = kb / 7, s = kb % 7;      // compile-time after unroll
        v2f a = *(const v2f*)(baseA + kb * 4); // ds_load_b64, imm offset

        v2f bb;                                 // imm-offset global loads
        bb.x = baseB[r * 64 * 448 + s * 64];
        bb.y = baseB[r * 64 * 448 + s * 64 + 32];

        if (kb & 1)
            acc1 = __builtin_amdgcn_wmma_f32_16x16x4_f32(
                false, a, false, bb, (short)0, acc1, false, false);
        else
            acc0 = __builtin_amdgcn_wmma_f32_16x16x4_f32(
                false, a, false, bb, (short)0, acc0, false, false);
    }

    v8f acc;
    #pragma unroll
    for (int i = 0; i < 8; ++i) acc[i] = acc0[i] + acc1[i];

    // Scatter D: VGPR v holds row m = v + 8*half; (kh,kw) = (m/4, m%4)
    float* ob = out + (size_t)((b * 3 + c) * 8 + t) * (128 * 128);
    #pragma unroll
    for (int v = 0; v < 8; ++v) {
        int mm = v + half * 8;
        int kh = mm >> 2, kw = mm & 3;
        ob[(kh * 32 + ay) * 128 + (kw * 32 + ax)] = acc[v];
    }
}

// ---------------------------------------------------------------------------
extern "C" void kernel_launch(void* const* d_in, const int* in_sizes, int n_in,
                              void* d_out, int out_size, void* d_ws, size_t ws_size,
                              hipStream_t stream) {
    const float* x      = (const float*)d_in[0];  // (2,3,8,448,448)
    const float* score  = (const float*)d_in[1];  // (2,4,196)
    const float* noise  = (const float*)d_in[2];  // (8,500,121)
    const float* sigma  = (const float*)d_in[3];  // scalar
    const int*   gid    = (const int*)d_in[4];    // (2,8)
    float* wreg = (float*)d_ws;                   // 8*121 floats scratch
    float* outp = (float*)d_out;                  // (2,3,8,128,128)

    weights_kernel<<<8, 128, 0, stream>>>(score, noise, sigma, wreg);
    gemm_kernel<<<384, 256, 0, stream>>>(x, gid, wreg, outp);
}